// WeightedMattingLoss_1932735283876
// MI455X (gfx1250) — compile-verified
//
#include <hip/hip_runtime.h>
#include <math.h>

#define BINS 10
#define TPB 256
#define NWAVE (TPB / 32)
#define EDGE 1.000001f     // 1.0 + 1e-6 in f32
#define TOT_HW 262144.0f   // 512*512 (reference uses H*W, not numel)
#define EPS_L1 1e-12f

typedef __attribute__((ext_vector_type(2))) float v2f;
typedef __attribute__((ext_vector_type(8))) float v8f;

// ---------------------------------------------------------------------------
// Wave-level sum via V_WMMA_F32_16X16X4_F32.
// A = ones(16x4), B holds one partial per lane (2nd slot zero).
// D[m][n] = sum_k B[k][n]; every row of D is the column-sum vector, so the
// grand total is invariant to the operand lane layout. VGPR0 of D holds the
// 16 column sums replicated in both 16-lane halves; finish with 4 immediate
// ds_swizzle XOR swaps (SWAPX8/4/2/1) -- no index setup, no bounds guards.
// Must be called with full-wave EXEC (uniform control flow).
// ---------------------------------------------------------------------------
__device__ __forceinline__ float wave_reduce_wmma(float x) {
  v2f a; a[0] = 1.0f; a[1] = 1.0f;
  v2f b; b[0] = x;    b[1] = 0.0f;
  v8f c = {};
  c = __builtin_amdgcn_wmma_f32_16x16x4_f32(
      /*neg_a=*/false, a, /*neg_b=*/false, b,
      /*c_mod=*/(short)0, c, /*reuse_a=*/false, /*reuse_b=*/false);
  float s = c[0];
  // group-of-32 swizzle: offset = {xor[14:10], or[9:5], and[4:0]}
  s += __int_as_float(__builtin_amdgcn_ds_swizzle(__float_as_int(s), 0x201F)); // SWAPX8
  s += __int_as_float(__builtin_amdgcn_ds_swizzle(__float_as_int(s), 0x101F)); // SWAPX4
  s += __int_as_float(__builtin_amdgcn_ds_swizzle(__float_as_int(s), 0x081F)); // SWAPX2
  s += __int_as_float(__builtin_amdgcn_ds_swizzle(__float_as_int(s), 0x041F)); // SWAPX1
  return s;  // total of all 32 lane inputs, in every lane
}

__device__ __forceinline__ int bin_of(float g) {
  int b = (int)(g * 10.0f);   // matches (g*BINS).astype(int32), g >= 0
  return b > (BINS - 1) ? (BINS - 1) : b;
}

// ---------------------------------------------------------------------------
// Zero the global histogram counters (2 pairs x 16-slot padded).
// ---------------------------------------------------------------------------
__global__ void ghm_init(unsigned* __restrict__ counts) {
  if (threadIdx.x < 32) counts[threadIdx.x] = 0u;
}

// ---------------------------------------------------------------------------
// Pass 1: histogram of g = |p - t| per pair. Per-wave LDS sub-histograms
// (ds_add atomics, one replica per wave to cut bank contention), merged with
// deterministic integer global atomics.
// ---------------------------------------------------------------------------
__global__ void ghm_hist(const float* __restrict__ pa, const float* __restrict__ ga,
                         const float* __restrict__ pc, const float* __restrict__ gc,
                         int n_a, int n_c, unsigned* __restrict__ counts) {
  __shared__ unsigned lh[NWAVE * 16];
  const int pair = blockIdx.y;
  const float* __restrict__ pf = pair ? pc : pa;
  const float* __restrict__ tf = pair ? gc : ga;
  const int n  = pair ? n_c : n_a;
  const int n4 = n >> 2;
  const float4* __restrict__ p = reinterpret_cast<const float4*>(pf);
  const float4* __restrict__ t = reinterpret_cast<const float4*>(tf);
  unsigned* __restrict__ gct = counts + pair * 16;
  unsigned* __restrict__ h = lh + ((threadIdx.x >> 5) << 4);

  for (int i = threadIdx.x; i < NWAVE * 16; i += TPB) lh[i] = 0u;
  __syncthreads();

  const int stride = gridDim.x * TPB;
  for (int i = blockIdx.x * TPB + threadIdx.x; i < n4; i += stride) {
    float4 pv = p[i];
    float4 tv = t[i];
    __builtin_prefetch(p + i + stride);   // global_prefetch_b8
    __builtin_prefetch(t + i + stride);
    float g;
    g = fabsf(pv.x - tv.x); if (g < EDGE) atomicAdd(&h[bin_of(g)], 1u);
    g = fabsf(pv.y - tv.y); if (g < EDGE) atomicAdd(&h[bin_of(g)], 1u);
    g = fabsf(pv.z - tv.z); if (g < EDGE) atomicAdd(&h[bin_of(g)], 1u);
    g = fabsf(pv.w - tv.w); if (g < EDGE) atomicAdd(&h[bin_of(g)], 1u);
  }
  // scalar tail (sizes here are multiples of 4; kept for generality)
  if (blockIdx.x == 0 && threadIdx.x == 0) {
    for (int j = n4 << 2; j < n; ++j) {
      float g = fabsf(pf[j] - tf[j]);
      if (g < EDGE) atomicAdd(&h[bin_of(g)], 1u);
    }
  }
  __syncthreads();
  if (threadIdx.x < BINS) {
    unsigned s = 0;
    #pragma unroll
    for (int w = 0; w < NWAVE; ++w) s += lh[(w << 4) + threadIdx.x];
    if (s) atomicAdd(&gct[threadIdx.x], s);
  }
}

// ---------------------------------------------------------------------------
// Pass 2: weighted L1 partial sums. Per-bin weights recomputed per block from
// finalized counts (10 loads). 2-way unrolled stream (independent
// accumulators for ILP), per-wave WMMA reduction, fixed-slot block partials.
// Pass 1 pulled both pairs (134 MB total) through L2 (192 MB) with RT hints,
// so this pass largely reads from L2.
// ---------------------------------------------------------------------------
__device__ __forceinline__ float term4(float4 pv, float4 tv,
                                       const float* __restrict__ wbin) {
  float acc = 0.0f, d, g, w;
  d = pv.x - tv.x; g = fabsf(d); w = (g < EDGE) ? wbin[bin_of(g)] : 0.0f;
  acc += sqrtf(w * (d * d) + EPS_L1);
  d = pv.y - tv.y; g = fabsf(d); w = (g < EDGE) ? wbin[bin_of(g)] : 0.0f;
  acc += sqrtf(w * (d * d) + EPS_L1);
  d = pv.z - tv.z; g = fabsf(d); w = (g < EDGE) ? wbin[bin_of(g)] : 0.0f;
  acc += sqrtf(w * (d * d) + EPS_L1);
  d = pv.w - tv.w; g = fabsf(d); w = (g < EDGE) ? wbin[bin_of(g)] : 0.0f;
  acc += sqrtf(w * (d * d) + EPS_L1);
  return acc;
}

__global__ void ghm_wsum(const float* __restrict__ pa, const float* __restrict__ ga,
                         const float* __restrict__ pc, const float* __restrict__ gc,
                         int n_a, int n_c,
                         const unsigned* __restrict__ counts,
                         float* __restrict__ partials, int nblk) {
  __shared__ float wbin[BINS];
  __shared__ float wavesum[NWAVE];
  const int pair = blockIdx.y;
  const float* __restrict__ pf = pair ? pc : pa;
  const float* __restrict__ tf = pair ? gc : ga;
  const int n  = pair ? n_c : n_a;
  const int n4 = n >> 2;
  const float4* __restrict__ p = reinterpret_cast<const float4*>(pf);
  const float4* __restrict__ t = reinterpret_cast<const float4*>(tf);

  if (threadIdx.x == 0) {
    const unsigned* c = counts + pair * 16;
    int nb = 0;
    #pragma unroll
    for (int i = 0; i < BINS; ++i) nb += (c[i] > 0u) ? 1 : 0;
    float inv_n = 1.0f / (float)(nb > 0 ? nb : 1);
    #pragma unroll
    for (int i = 0; i < BINS; ++i)
      wbin[i] = (c[i] > 0u) ? (TOT_HW / (0.9f * (float)c[i])) * inv_n : 0.0f;
  }
  __syncthreads();

  float acc0 = 0.0f, acc1 = 0.0f;
  const int stride = gridDim.x * TPB;
  int i = blockIdx.x * TPB + threadIdx.x;
  for (; i + stride < n4; i += 2 * stride) {
    float4 pv0 = p[i];
    float4 tv0 = t[i];
    float4 pv1 = p[i + stride];
    float4 tv1 = t[i + stride];
    __builtin_prefetch(p + i + 2 * stride);
    __builtin_prefetch(t + i + 2 * stride);
    acc0 += term4(pv0, tv0, wbin);
    acc1 += term4(pv1, tv1, wbin);
  }
  if (i < n4) acc0 += term4(p[i], t[i], wbin);
  if (blockIdx.x == 0 && threadIdx.x == 0) {
    for (int j = n4 << 2; j < n; ++j) {
      float d = pf[j] - tf[j];
      float g = fabsf(d);
      float w = (g < EDGE) ? wbin[bin_of(g)] : 0.0f;
      acc1 += sqrtf(w * (d * d) + EPS_L1);
    }
  }

  // uniform control flow from here: full-wave EXEC for WMMA
  float s = wave_reduce_wmma(acc0 + acc1);
  if ((threadIdx.x & 31) == 0) wavesum[threadIdx.x >> 5] = s;
  __syncthreads();
  if (threadIdx.x == 0) {
    float b = 0.0f;
    #pragma unroll
    for (int w = 0; w < NWAVE; ++w) b += wavesum[w];  // fixed order
    partials[pair * nblk + blockIdx.x] = b;
  }
}

// ---------------------------------------------------------------------------
// Final: reduce per-block partials for each pair, emit 3 scalars.
// ---------------------------------------------------------------------------
__global__ void ghm_final(const float* __restrict__ partials, int nblk,
                          float* __restrict__ out, int n_a, int n_c) {
  __shared__ float wavesum[NWAVE];
  __shared__ float pairsum[2];
  for (int pair = 0; pair < 2; ++pair) {
    float acc = 0.0f;
    for (int i = threadIdx.x; i < nblk; i += TPB)
      acc += partials[pair * nblk + i];     // deterministic per-thread order
    float s = wave_reduce_wmma(acc);
    if ((threadIdx.x & 31) == 0) wavesum[threadIdx.x >> 5] = s;
    __syncthreads();
    if (threadIdx.x == 0) {
      float t = 0.0f;
      #pragma unroll
      for (int w = 0; w < NWAVE; ++w) t += wavesum[w];
      pairsum[pair] = t;
    }
    __syncthreads();
  }
  if (threadIdx.x == 0) {
    float a = pairsum[0] / (float)n_a;
    float c = pairsum[1] / (float)n_c;
    out[0] = 0.5f * (a + c);
    out[1] = a;
    out[2] = c;
  }
}

extern "C" void kernel_launch(void* const* d_in, const int* in_sizes, int n_in,
                              void* d_out, int out_size, void* d_ws, size_t ws_size,
                              hipStream_t stream) {
  const float* pa = (const float*)d_in[0];  // pred_alphas
  const float* ga = (const float*)d_in[1];  // gt_alphas
  const float* pc = (const float*)d_in[2];  // pred_comps
  const float* gc = (const float*)d_in[3];  // gt_comps
  const int n_a = in_sizes[0];              // 16*1*512*512
  const int n_c = in_sizes[2];              // 16*3*512*512

  unsigned* counts = (unsigned*)d_ws;                       // 32 u32 (128 B)
  float* partials  = (float*)((char*)d_ws + 32 * sizeof(unsigned));

  int nblk = 2048;
  while (nblk > 1 &&
         (size_t)(32 * sizeof(unsigned) + 2 * (size_t)nblk * sizeof(float)) > ws_size)
    nblk >>= 1;

  ghm_init<<<dim3(1), dim3(32), 0, stream>>>(counts);
  dim3 grid(nblk, 2);
  ghm_hist<<<grid, dim3(TPB), 0, stream>>>(pa, ga, pc, gc, n_a, n_c, counts);
  ghm_wsum<<<grid, dim3(TPB), 0, stream>>>(pa, ga, pc, gc, n_a, n_c,
                                           counts, partials, nblk);
  ghm_final<<<dim3(1), dim3(TPB), 0, stream>>>(partials, nblk, (float*)d_out,
                                               n_a, n_c);
}